// MyNCALoss_57509612094150
// MI455X (gfx1250) — compile-verified
//
#include <hip/hip_runtime.h>
#include <math.h>

typedef __attribute__((ext_vector_type(2))) float v2f;
typedef __attribute__((ext_vector_type(8))) float v8f;

#define DIM 1024
#define SOFTMAX_SCALE 3.5f

// One wave (32 lanes) handles a 16-row tile. Cross dots a.p / a.n via
// V_WMMA_F32_16X16X4_F32 diagonals; self dots (norms) via VALU FMA.
__global__ __launch_bounds__(256) void nca_tile_kernel(
    const float* __restrict__ A, const float* __restrict__ P,
    const float* __restrict__ N, float* __restrict__ ws, int ntiles)
{
    const int lane = threadIdx.x & 31;
    const int wave = threadIdx.x >> 5;
    const int tile = blockIdx.x * (blockDim.x >> 5) + wave;
    if (tile >= ntiles) return;

    // f32 WMMA A(16x4) layout: lane L (L<16) holds row L, K={k0,k0+1};
    // lane L+16 holds row L, K={k0+2,k0+3}. B(4x16) layout: lane n holds
    // column n (= row n of P/N tile), same K split -> identical addressing.
    const int rowInTile = lane & 15;
    const int khalf     = (lane >> 4) << 1;            // 0 or 2
    const size_t base   = (size_t)(tile * 16 + rowInTile) * DIM + khalf;
    const float2* pa = (const float2*)(A + base);
    const float2* pp = (const float2*)(P + base);
    const float2* pn = (const float2*)(N + base);

    v8f cAP = {};  // C[m][n] accumulates dot(a_m, p_n) over K
    v8f cAN = {};
    float aa = 0.f, pq = 0.f, nq = 0.f;

#pragma unroll 8
    for (int k0 = 0; k0 < DIM; k0 += 4) {
        const int i2 = k0 >> 1;                        // float2 index
        float2 av = pa[i2];
        float2 pv = pp[i2];
        float2 nv = pn[i2];
        v2f aA; aA[0] = av.x; aA[1] = av.y;
        v2f pB; pB[0] = pv.x; pB[1] = pv.y;
        v2f nB; nB[0] = nv.x; nB[1] = nv.y;
        cAP = __builtin_amdgcn_wmma_f32_16x16x4_f32(
                  false, aA, false, pB, (short)0, cAP, false, false);
        cAN = __builtin_amdgcn_wmma_f32_16x16x4_f32(
                  false, aA, false, nB, (short)0, cAN, false, false);
        aa = fmaf(av.x, av.x, fmaf(av.y, av.y, aa));
        pq = fmaf(pv.x, pv.x, fmaf(pv.y, pv.y, pq));
        nq = fmaf(nv.x, nv.x, fmaf(nv.y, nv.y, nq));
    }

    // Fold the K-half partner lane (lane ^ 16): full-row sums of squares.
    // After this, lane L holds norms for row (L & 15).
    aa += __shfl_xor(aa, 16, 32);
    pq += __shfl_xor(pq, 16, 32);
    nq += __shfl_xor(nq, 16, 32);

    // Diagonal of C: row m in [0,8)  -> lane m,    component m
    //               row m in [8,16) -> lane m+16, component m-8
    const bool active = (lane < 8) || (lane >= 24);
    const int  m = lane & 7;
    float dAP = cAP[0], dAN = cAN[0];
#pragma unroll
    for (int i = 1; i < 8; ++i) {
        if (m == i) { dAP = cAP[i]; dAN = cAN[i]; }
    }

    if (active) {
        const int r = (lane < 8) ? lane : (lane - 16);   // == lane & 15
        float na = fmaxf(sqrtf(aa), 1e-12f);
        float np = fmaxf(sqrtf(pq), 1e-12f);
        float nn = fmaxf(sqrtf(nq), 1e-12f);
        float sap = dAP / (na * np);
        float san = dAN / (na * nn);
        float z = SOFTMAX_SCALE * (san - sap);
        // logaddexp(0, z) = max(z,0) + log1p(exp(-|z|))
        float loss = fmaxf(z, 0.f) + log1pf(expf(-fabsf(z)));
        ws[tile * 16 + r] = loss;
    }
}

// Deterministic single-block tree reduction: mean over n per-row losses.
__global__ __launch_bounds__(256) void nca_reduce_kernel(
    const float* __restrict__ ws, float* __restrict__ out, int n)
{
    __shared__ float smem[256];
    float acc = 0.f;
    for (int i = threadIdx.x; i < n; i += 256) acc += ws[i];
    smem[threadIdx.x] = acc;
    __syncthreads();
#pragma unroll
    for (int off = 128; off > 0; off >>= 1) {
        if ((int)threadIdx.x < off) smem[threadIdx.x] += smem[threadIdx.x + off];
        __syncthreads();
    }
    if (threadIdx.x == 0) out[0] = smem[0] * (1.0f / (float)n);
}

extern "C" void kernel_launch(void* const* d_in, const int* in_sizes, int n_in,
                              void* d_out, int out_size, void* d_ws, size_t ws_size,
                              hipStream_t stream) {
    const float* A = (const float*)d_in[0];
    const float* P = (const float*)d_in[1];
    const float* N = (const float*)d_in[2];
    float* ws  = (float*)d_ws;
    float* out = (float*)d_out;

    const int B = in_sizes[0] / DIM;       // 16384
    const int ntiles = B / 16;             // 1024 tiles, one wave each

    const int wavesPerBlock = 8;           // 256 threads = 8 wave32
    const int grid = (ntiles + wavesPerBlock - 1) / wavesPerBlock;

    nca_tile_kernel<<<grid, 256, 0, stream>>>(A, P, N, ws, ntiles);
    nca_reduce_kernel<<<1, 256, 0, stream>>>(ws, out, B);
}